// Quadratic_50663434223733
// MI455X (gfx1250) — compile-verified
//
#include <hip/hip_runtime.h>

typedef __attribute__((ext_vector_type(2))) float v2f;
typedef __attribute__((ext_vector_type(4))) float v4f;
typedef __attribute__((ext_vector_type(8))) float v8f;

#define DIM 1024
#define NTILES (DIM / 16)            // 64
#define ROWS_PER_WG 64               // 64 x-rows staged in LDS per WG
#define NWAVES 8                     // 4 waves share each 32-row block, split N tiles 4-way
#define XS_STRIDE (DIM + 4)          // pad 4 floats: conflict-free A reads, b64 aligned

// Packed-B float offset of tile nt: 16 floats per (k,n-tile) row, k from n0=16*nt.
__host__ __device__ inline size_t pk_off(int nt) {
    return (size_t)16384 * nt - (size_t)128 * nt * (nt - 1);
}

// ---------------------------------------------------------------------------
// Kernel 1: build packed WMMA B-fragments directly from packed tril L_vars.
// Per N-tile nt, groups of 2 WMMAs (K depth 8); lane l (m=l&15, hi=l>>4) owns
// 4 consecutive dwords: L[kk+2hi][n0+m], L[kk+2hi+1][n0+m],
//                       L[kk+4+2hi][n0+m], L[kk+4+2hi+1][n0+m]
// ---------------------------------------------------------------------------
__global__ void build_packed_B(const float* __restrict__ Lv, float* __restrict__ pk) {
    const int nt = blockIdx.x;                 // 0..63
    const int n0 = nt * 16;
    const size_t base = pk_off(nt);
    const int nelem = 16 * (DIM - n0);
    for (int i = threadIdx.x; i < nelem; i += blockDim.x) {
        const int g    = i >> 7;               // group (2 WMMAs, K depth 8)
        const int r    = i & 127;
        const int lane = r >> 2;
        const int dw   = r & 3;
        const int m    = lane & 15;
        const int hi   = lane >> 4;
        const int row  = n0 + 8 * g + 2 * hi + (dw & 1) + 4 * (dw >> 1);  // K index
        const int col  = n0 + m;                                          // N index
        float v = 0.0f;
        if (col <= row) v = Lv[((row * (row + 1)) >> 1) + col];
        pk[base + i] = v;
    }
}

// ---------------------------------------------------------------------------
// Kernel 2: y[i] = ||x_i @ L||^2 + x_i.b + c  via V_WMMA_F32_16X16X4_F32.
// Wave w: rows block half=w>>2 (32 rows = 2 M-tiles), quarter q=w&3.
// Tiles assigned snake-wise (q, 7-q per octet) -> perfectly balanced K work.
// Each B fragment feeds TWO WMMAs (both M-tiles) -> half the L2 B traffic and
// two independent accumulator chains.
// ---------------------------------------------------------------------------
__global__ __launch_bounds__(NWAVES * 32)
void quad_wmma_kernel(const float* __restrict__ X,
                      const float* __restrict__ pk,
                      const float* __restrict__ bvec,
                      const float* __restrict__ cscalar,
                      float* __restrict__ out) {
    __shared__ float Xs[ROWS_PER_WG * XS_STRIDE];   // ~257 KB
    __shared__ float xb_s[ROWS_PER_WG];
    __shared__ float quad_s[ROWS_PER_WG];

    const int tid  = threadIdx.x;
    const int lane = tid & 31;
    const int wave = __builtin_amdgcn_readfirstlane(tid >> 5);  // scalar wave id
    const int wg_row0 = blockIdx.x * ROWS_PER_WG;

    if (tid < ROWS_PER_WG) quad_s[tid] = 0.0f;

    // ---- Stage x: wave w stages rows [8w, 8w+8) (coalesced along k) ----
    for (int j = 0; j < 8; ++j) {
        const int r = wave * 8 + j;
        const float* src = X + (size_t)(wg_row0 + r) * DIM;
        float* dst = Xs + r * XS_STRIDE;
        for (int k = lane; k < DIM; k += 32) dst[k] = src[k];
    }

    __syncthreads();

    // ---- x.b for this wave's 8 staged rows (cooperative wave reduction) ----
    for (int j = 0; j < 8; ++j) {
        const int r = wave * 8 + j;
        const float* xr = Xs + r * XS_STRIDE;
        float p = 0.0f;
        for (int k = lane; k < DIM; k += 32) p += xr[k] * bvec[k];
        for (int off = 16; off >= 1; off >>= 1) p += __shfl_xor(p, off, 32);
        if (lane == 0) xb_s[r] = p;
    }

    // ---- WMMA main loop ----
    // C/D layout: VGPR v: lanes 0-15 -> (M=v, N=lane); lanes 16-31 -> (M=v+8, N=lane-16)
    const int m    = lane & 15;
    const int hi   = lane >> 4;
    const int koff = hi * 2;
    const int half = wave >> 2;            // which 32-row block (scalar)
    const int q    = wave & 3;             // N-tile quarter (scalar)

    const float* arow0 = Xs + (half * 32 + m) * XS_STRIDE;        // M-tile A
    const float* arow1 = Xs + (half * 32 + 16 + m) * XS_STRIDE;   // M-tile B

    float qA[8], qB[8];
#pragma unroll
    for (int v = 0; v < 8; ++v) { qA[v] = 0.0f; qB[v] = 0.0f; }

    for (int i = 0; i < NTILES / 4; ++i) {
        const int oct = i >> 1;
        const int nt  = oct * 8 + ((i & 1) ? (7 - q) : q);   // snake: balanced K work
        const int n0  = nt * 16;

        v8f accA = {0.0f, 0.0f, 0.0f, 0.0f, 0.0f, 0.0f, 0.0f, 0.0f};
        v8f accB = {0.0f, 0.0f, 0.0f, 0.0f, 0.0f, 0.0f, 0.0f, 0.0f};

        const float* ap0 = arow0 + n0 + koff;
        const float* ap1 = arow1 + n0 + koff;
        const float* pkp = pk + pk_off(nt) + (size_t)lane * 4;
        const int groups = 128 - 2 * nt;                     // scalar trip count

#pragma unroll 2
        for (int it = 0; it < groups; ++it) {
            v4f bq  = *(const v4f*)pkp;                      // b128: B for 2 K-steps
            v2f a00 = *(const v2f*)ap0;                      // ds b64 pairs
            v2f a01 = *(const v2f*)(ap0 + 4);
            v2f a10 = *(const v2f*)ap1;
            v2f a11 = *(const v2f*)(ap1 + 4);
            v2f b0  = {bq.x, bq.y};
            v2f b1  = {bq.z, bq.w};
            accA = __builtin_amdgcn_wmma_f32_16x16x4_f32(
                false, a00, false, b0, (short)0, accA, false, false);
            accB = __builtin_amdgcn_wmma_f32_16x16x4_f32(
                false, a10, false, b0, (short)0, accB, false, false);
            accA = __builtin_amdgcn_wmma_f32_16x16x4_f32(
                false, a01, false, b1, (short)0, accA, false, false);
            accB = __builtin_amdgcn_wmma_f32_16x16x4_f32(
                false, a11, false, b1, (short)0, accB, false, false);
            ap0 += 8;
            ap1 += 8;
            pkp += 128;
        }

#pragma unroll
        for (int v = 0; v < 8; ++v) {
            qA[v] += accA[v] * accA[v];
            qB[v] += accB[v] * accB[v];
        }
    }

    // ---- reduce quad partials across the 16 lanes of each half ----
#pragma unroll
    for (int off = 8; off >= 1; off >>= 1) {
#pragma unroll
        for (int v = 0; v < 8; ++v) {
            qA[v] += __shfl_xor(qA[v], off, 32);
            qB[v] += __shfl_xor(qB[v], off, 32);
        }
    }

    // lanes 0 / 16 hold rows 0-7 / 8-15 of each M-tile; merge 4 quarter-waves
    if (m == 0) {
        const int rbaseA = half * 32 + hi * 8;
        const int rbaseB = half * 32 + 16 + hi * 8;
#pragma unroll
        for (int v = 0; v < 8; ++v) {
            atomicAdd(&quad_s[rbaseA + v], qA[v]);
            atomicAdd(&quad_s[rbaseB + v], qB[v]);
        }
    }

    __syncthreads();

    // ---- final: y = quad + x.b + c ----
    if (tid < ROWS_PER_WG) {
        out[wg_row0 + tid] = quad_s[tid] + xb_s[tid] + cscalar[0];
    }
}

// ---------------------------------------------------------------------------
extern "C" void kernel_launch(void* const* d_in, const int* in_sizes, int n_in,
                              void* d_out, int out_size, void* d_ws, size_t ws_size,
                              hipStream_t stream) {
    const float* x  = (const float*)d_in[0];
    const float* Lv = (const float*)d_in[1];
    const float* bv = (const float*)d_in[2];
    const float* cv = (const float*)d_in[3];
    float* out = (float*)d_out;
    float* pk  = (float*)d_ws;            // packed B fragments: 532480 floats (~2.03 MB)

    const int Brows = in_sizes[0] / DIM;  // 65536

    build_packed_B<<<NTILES, 256, 0, stream>>>(Lv, pk);
    quad_wmma_kernel<<<Brows / ROWS_PER_WG, NWAVES * 32, 0, stream>>>(x, pk, bv, cv, out);
}